// AffineQuantizedKVCache_50105088475452
// MI455X (gfx1250) — compile-verified
//
#include <hip/hip_runtime.h>
#include <stdint.h>

// ---------------------------------------------------------------------------
// AffineQuantizedKVCache update+dequant for MI455X (gfx1250).
//
// Reference semantics reduce to:
//   out_k[b,h,s,:] = float(k_cache[b,h,s,:]) * k_scale[b,h,s]   (all s)
//   out_k[b,h,input_pos[i],:] = k_val[b,h,i,:]                  (fresh rows)
//   (same for V; outputs concatenated: K first, then V)
// The int8 quantization in the reference never reaches the outputs.
//
// Memory-bound streaming kernel: ~1 GB traffic @ 23.3 TB/s => ~45us floor.
// CDNA5 features: async global->LDS staging (ASYNCcnt, double-buffered),
// s_wait_asynccnt, b128 accesses, NT temporal hints (working set >> 192MB L2).
// K/V tensor selection is uniform per block (blockIdx.y) so base pointers
// stay in SGPRs and per-lane address math is 32-bit only.
// ---------------------------------------------------------------------------

typedef int   __attribute__((ext_vector_type(4))) i32x4;
typedef float __attribute__((ext_vector_type(4))) f32x4;

// Address-space-qualified views of a 16B chunk for the async-LDS builtin,
// whose params are pointers to <4 x i32> (per compile-probe diagnostics).
typedef __attribute__((address_space(1))) i32x4 as1_i32x4;
typedef __attribute__((address_space(3))) i32x4 as3_i32x4;

#define B_  4
#define H_  32
#define S_  4096
#define D_  128
#define SN_ 16

static constexpr long NK = (long)B_ * H_ * S_ * D_;    // 67,108,864 elems/tensor
static constexpr long NF = (long)B_ * H_ * SN_ * D_;   // 262,144 fresh elems/tensor

#if defined(__has_builtin)
#if __has_builtin(__builtin_amdgcn_global_load_async_to_lds_b128) && \
    __has_builtin(__builtin_amdgcn_s_wait_asynccnt)
#define USE_ASYNC 1
#endif
#endif
#ifndef USE_ASYNC
#define USE_ASYNC 0
#pragma message("gfx1250 async-to-LDS builtins NOT found; compiling direct-load fallback")
#endif

#if USE_ASYNC
__device__ __forceinline__ void stage16(const int* gsrc, void* lds_slot) {
  // 16B/lane async copy global -> LDS; tracked by ASYNCcnt. cpol=1 => TH_NT
  // (read-once stream, keep L2 clean). Low 32 bits of a flat LDS address are
  // the LDS byte offset (aperture rule), so the truncating cast is correct.
  __builtin_amdgcn_global_load_async_to_lds_b128(
      (as1_i32x4*)(uintptr_t)gsrc,
      (as3_i32x4*)(uint32_t)(uintptr_t)lds_slot,
      /*offset=*/0, /*cpol=*/1);
}
#endif

// -------- Kernel 1: full-cache dequant, one tensor per blockIdx.y -----------
__global__ __launch_bounds__(256)
void kvcache_dequant_kernel(const int* __restrict__ kc, const int* __restrict__ vc,
                            const float* __restrict__ ks, const float* __restrict__ vs,
                            float* __restrict__ out) {
  constexpr int ITERS = 8;
  const int tid = threadIdx.x;
  // Uniform (scalar) tensor select: y==0 -> K, y==1 -> V.
  const int        y    = blockIdx.y;
  const int*  __restrict__ src = y ? vc : kc;
  const float* __restrict__ sc = y ? vs : ks;
  float*      __restrict__ ob  = out + (y ? NK : 0);

  const unsigned t      = blockIdx.x * 256u + (unsigned)tid;  // group id
  const unsigned stride = gridDim.x * 256u;                   // groups/sweep

#if USE_ASYNC
  __shared__ i32x4 sbuf[2][256];                              // wave-private, 8KB
  stage16(src + ((size_t)t << 2), &sbuf[0][tid]);
#pragma unroll
  for (int it = 0; it < ITERS; ++it) {
    const unsigned g = t + (unsigned)it * stride;             // < 16M
    if (it + 1 < ITERS) {
      stage16(src + ((size_t)(g + stride) << 2), &sbuf[(it + 1) & 1][tid]);
      __builtin_amdgcn_s_wait_asynccnt(1);  // current buffer landed in LDS
    } else {
      __builtin_amdgcn_s_wait_asynccnt(0);
    }
    i32x4 q = sbuf[it & 1][tid];
    float s = sc[g >> 5];                   // one scale per 128-elem row
    f32x4 r;
    r.x = (float)q.x * s;
    r.y = (float)q.y * s;
    r.z = (float)q.z * s;
    r.w = (float)q.w * s;
    __builtin_nontemporal_store(r, (f32x4*)(ob + ((size_t)g << 2)));
  }
#else
#pragma unroll
  for (int it = 0; it < ITERS; ++it) {
    const unsigned g = t + (unsigned)it * stride;
    i32x4 q = __builtin_nontemporal_load((const i32x4*)(src + ((size_t)g << 2)));
    float s = sc[g >> 5];
    f32x4 r;
    r.x = (float)q.x * s;
    r.y = (float)q.y * s;
    r.z = (float)q.z * s;
    r.w = (float)q.w * s;
    __builtin_nontemporal_store(r, (f32x4*)(ob + ((size_t)g << 2)));
  }
#endif
}

// -------- Kernel 2: overwrite the 16 fresh rows with exact fp32 values ------
__global__ __launch_bounds__(256)
void kvcache_scatter_kernel(const float* __restrict__ kv, const float* __restrict__ vv,
                            const int* __restrict__ pos, float* __restrict__ out) {
  const long t  = (long)blockIdx.x * 256 + threadIdx.x;  // group id over 2*NF/4
  const long e  = t << 2;
  const bool isV = (e >= NF);
  const long le  = isV ? (e - NF) : e;
  const float* src = isV ? vv : kv;

  f32x4 val = *(const f32x4*)(src + le);

  const int  d   = (int)(le & (D_ - 1));  // element within row (4-aligned)
  const long row = le >> 7;               // b*H*SN + h*SN + i
  const int  i   = (int)(row & (SN_ - 1));
  const long bh  = row >> 4;              // b*H + h
  const int  s   = pos[i];
  const long orow = bh * (long)S_ + (long)s;

  float* dst = out + (isV ? NK : 0) + (orow << 7) + d;
  __builtin_nontemporal_store(val, (f32x4*)dst);
}

extern "C" void kernel_launch(void* const* d_in, const int* in_sizes, int n_in,
                              void* d_out, int out_size, void* d_ws, size_t ws_size,
                              hipStream_t stream) {
  (void)in_sizes; (void)n_in; (void)out_size; (void)d_ws; (void)ws_size;
  // setup_inputs() dict order:
  const int*   pos = (const int*)  d_in[0];   // [16]
  const float* kv  = (const float*)d_in[1];   // [4,32,16,128]
  const float* vv  = (const float*)d_in[2];   // [4,32,16,128]
  const int*   kc  = (const int*)  d_in[3];   // [4,32,4096,128] (int8 values, int32 storage)
  const int*   vc  = (const int*)  d_in[4];   // [4,32,4096,128]
  const float* ks  = (const float*)d_in[5];   // [4,32,4096,1]
  const float* vs  = (const float*)d_in[6];   // [4,32,4096,1]
  float*       out = (float*)d_out;           // [k_out | v_out], 2*NK floats

  // Per tensor: NK/4 = 16M groups; 256 threads * 8 iters per thread
  // -> gridDim.x = 16M / (256*8) = 8192; gridDim.y selects K or V.
  const unsigned gx = (unsigned)((NK / 4) / (256 * 8));
  dim3 grid1(gx, 2, 1);
  kvcache_dequant_kernel<<<grid1, 256, 0, stream>>>(kc, vc, ks, vs, out);

  // 2*NF elements / 4-per-thread / 256 threads = 512 blocks
  const unsigned grid2 = (unsigned)((2 * NF) / 4 / 256);
  kvcache_scatter_kernel<<<grid2, 256, 0, stream>>>(kv, vv, pos, out);
}